// Model_67293547594179
// MI455X (gfx1250) — compile-verified
//
#include <hip/hip_runtime.h>
#include <hip/hip_bf16.h>
#include <stdint.h>

// moe_align_block_size for MI455X (gfx1250, wave32).
static constexpr int E       = 64;   // num_experts (fixed by setup_inputs)
static constexpr int MOE_B   = 128;  // block_size  (fixed by setup_inputs)
static constexpr int THREADS = 256;
static constexpr int WAVES   = THREADS / 32;
static constexpr int DEPTH   = 4;    // async pipeline depth (tiles in flight)
static constexpr int TILE    = 128;  // ints per async tile = 512B (b128 x 32 lanes)

// ---------------- CDNA5 async global->LDS helpers (inline asm) ----------------
__device__ __forceinline__ void async_load_b128(unsigned lds_byte_addr,
                                                unsigned src_byte_off,
                                                unsigned long long base) {
  asm volatile("global_load_async_to_lds_b128 %0, %1, %2"
               :: "v"(lds_byte_addr), "v"(src_byte_off), "s"(base)
               : "memory");
}
__device__ __forceinline__ void wait_async(int n) {
  switch (n) {
    case 0:  asm volatile("s_wait_asynccnt 0x0" ::: "memory"); break;
    case 1:  asm volatile("s_wait_asynccnt 0x1" ::: "memory"); break;
    case 2:  asm volatile("s_wait_asynccnt 0x2" ::: "memory"); break;
    default: asm volatile("s_wait_asynccnt 0x3" ::: "memory"); break;
  }
}
__device__ __forceinline__ void wait_ds0() {
  asm volatile("s_wait_dscnt 0x0" ::: "memory");
}
// Low 32 bits of a flat LDS pointer == LDS byte offset (ISA 10.2).
__device__ __forceinline__ unsigned lds_off(const void* p) {
  return (unsigned)(uintptr_t)p;
}
__device__ __forceinline__ int imin(int a, int b) { return a < b ? a : b; }

// ---------------- Stage 1: per-chunk histogram (async-LDS streaming) ----------------
// chunk_counts layout: TRANSPOSED [E][num_chunks] for coalesced per-expert scans.
__global__ void k_hist(const int* __restrict__ flat, int numel, int chunk,
                       int num_chunks, int* __restrict__ chunk_counts) {
  __shared__ __align__(16) int stage[WAVES][DEPTH][TILE];
  __shared__ int hist[E];
  const int tid = threadIdx.x, wave = tid >> 5, lane = tid & 31;

  for (int e = tid; e < E; e += THREADS) hist[e] = 0;
  __syncthreads();

  const long long cbase = (long long)blockIdx.x * chunk;
  const int wints  = chunk / WAVES;     // ints per wave slice (multiple of 128)
  const int ntiles = wints / TILE;
  const unsigned long long wsrc =
      (unsigned long long)(flat + cbase + (long long)wave * wints);
  const bool full = (cbase + (long long)(wave + 1) * wints) <= (long long)numel;

  if (full) {
    unsigned ldsb[DEPTH];
    #pragma unroll
    for (int b = 0; b < DEPTH; ++b) ldsb[b] = lds_off(&stage[wave][b][lane * 4]);
    const int k0 = imin(DEPTH, ntiles);
    for (int k = 0; k < k0; ++k)
      async_load_b128(ldsb[k], (unsigned)(k * 512 + lane * 16), wsrc);
    for (int t = 0; t < ntiles; ++t) {
      wait_async(imin(DEPTH - 1, ntiles - 1 - t));
      const int buf = t & (DEPTH - 1);
      const int4 v = *reinterpret_cast<const int4*>(&stage[wave][buf][lane * 4]);
      wait_ds0();  // values in regs; safe to overwrite this buffer
      if (t + DEPTH < ntiles)
        async_load_b128(ldsb[buf], (unsigned)((t + DEPTH) * 512 + lane * 16), wsrc);
      atomicAdd(&hist[v.x], 1);
      atomicAdd(&hist[v.y], 1);
      atomicAdd(&hist[v.z], 1);
      atomicAdd(&hist[v.w], 1);
    }
  } else {
    for (int i = lane; i < wints; i += 32) {
      const long long g = cbase + (long long)wave * wints + i;
      if (g < numel) atomicAdd(&hist[flat[g]], 1);
    }
  }
  __syncthreads();
  for (int e = tid; e < E; e += THREADS)
    chunk_counts[(size_t)e * num_chunks + blockIdx.x] = hist[e];
}

// ---------------- Stage 2a: parallel exclusive scan per expert ----------------
// One block per expert; coalesced row scan over num_chunks entries (in place).
__global__ void k_scan_expert(int* __restrict__ cc, int num_chunks,
                              int* __restrict__ counts) {
  __shared__ int tsum[THREADS];
  int* row = cc + (size_t)blockIdx.x * num_chunks;
  const int per   = (num_chunks + THREADS - 1) / THREADS;
  const int begin = threadIdx.x * per;
  const int end   = imin(begin + per, num_chunks);

  int local = 0;
  for (int i = begin; i < end; ++i) local += row[i];
  tsum[threadIdx.x] = local;
  __syncthreads();
  // Hillis-Steele inclusive scan over 256 partials
  for (int off = 1; off < THREADS; off <<= 1) {
    const int v = (threadIdx.x >= off) ? tsum[threadIdx.x - off] : 0;
    __syncthreads();
    tsum[threadIdx.x] += v;
    __syncthreads();
  }
  int run = (threadIdx.x == 0) ? 0 : tsum[threadIdx.x - 1];  // exclusive base
  for (int i = begin; i < end; ++i) { const int t = row[i]; row[i] = run; run += t; }
  if (threadIdx.x == THREADS - 1) counts[blockIdx.x] = tsum[THREADS - 1];
}

// ---------------- Stage 2b: block-padded cumsum over 64 experts ----------------
__global__ void k_cum(const int* __restrict__ counts, int* __restrict__ cum,
                      int* __restrict__ out_total) {
  __shared__ int cumS[E + 1];
  if (threadIdx.x == 0) {
    int acc = 0;
    cumS[0] = 0;
    for (int j = 0; j < E; ++j) {
      acc += ((counts[j] + MOE_B - 1) / MOE_B) * MOE_B;
      cumS[j + 1] = acc;
    }
  }
  __syncthreads();
  if (threadIdx.x <= E) cum[threadIdx.x] = cumS[threadIdx.x];
  if (threadIdx.x == 0) out_total[0] = cumS[E];
}

// ---------------- Stage 3: fill only padding gaps with `numel` ----------------
__global__ void k_fill(int* __restrict__ sorted, const int* __restrict__ cum,
                       const int* __restrict__ counts, int numel, int n1) {
  const int e = blockIdx.x;  // 0..E
  int start, end;
  if (e < E) { start = cum[e] + counts[e]; end = cum[e + 1]; }
  else       { start = cum[E];             end = n1; }
  for (int i = start + (int)threadIdx.x; i < end; i += (int)blockDim.x)
    sorted[i] = numel;
}

// ---------------- Stage 4: expert id per M-block (searchsorted in LDS) ----------------
__global__ void k_expert_ids(int* __restrict__ expert_ids,
                             const int* __restrict__ cum, int nblocks) {
  __shared__ int cumS[E + 1];
  if (threadIdx.x <= E) cumS[threadIdx.x] = cum[threadIdx.x];
  __syncthreads();
  const int b = blockIdx.x * blockDim.x + threadIdx.x;
  if (b >= nblocks) return;
  const int bstart = b * MOE_B;
  int eid = 0;
  #pragma unroll
  for (int j = 1; j <= E; ++j) eid += (cumS[j] <= bstart) ? 1 : 0;
  expert_ids[b] = (bstart < cumS[E]) ? eid : 0;
}

// ---------------- Stage 5: stable scatter (one chunk per wave, async-staged) ----------------
__global__ void k_scatter(const int* __restrict__ flat, int numel, int chunk,
                          int num_chunks,
                          const int* __restrict__ chunk_excl,  // [E][num_chunks] exclusive
                          const int* __restrict__ cum,
                          int* __restrict__ sorted) {
  __shared__ __align__(16) int stage[WAVES][DEPTH][TILE];
  __shared__ int offs[WAVES][E];
  const int wave = threadIdx.x >> 5, lane = threadIdx.x & 31;
  const int c = blockIdx.x * WAVES + wave;
  const bool live = (c < num_chunks);
  if (live)
    for (int e = lane; e < E; e += 32)
      offs[wave][e] = cum[e] + chunk_excl[(size_t)e * num_chunks + c];
  __syncthreads();
  if (!live) return;

  const long long base = (long long)c * chunk;
  const unsigned lt = (1u << lane) - 1u;
  const bool full = (base + chunk) <= (long long)numel;

  if (full) {
    const int ntiles = chunk / TILE;
    const unsigned long long wsrc = (unsigned long long)(flat + base);
    unsigned ldsb[DEPTH];
    #pragma unroll
    for (int b = 0; b < DEPTH; ++b) ldsb[b] = lds_off(&stage[wave][b][lane * 4]);
    const int k0 = imin(DEPTH, ntiles);
    for (int k = 0; k < k0; ++k)
      async_load_b128(ldsb[k], (unsigned)(k * 512 + lane * 16), wsrc);

    for (int t = 0; t < ntiles; ++t) {
      wait_async(imin(DEPTH - 1, ntiles - 1 - t));
      const int buf = t & (DEPTH - 1);
      // read tile in FLAT order: sub-step j handles elements t*128 + j*32 + lane
      int ev[4];
      #pragma unroll
      for (int j = 0; j < 4; ++j) ev[j] = stage[wave][buf][j * 32 + lane];
      wait_ds0();  // regs hold data; buffer reusable
      if (t + DEPTH < ntiles)
        async_load_b128(ldsb[buf], (unsigned)((t + DEPTH) * 512 + lane * 16), wsrc);

      #pragma unroll
      for (int j = 0; j < 4; ++j) {
        const int e = ev[j];
        unsigned mask = 0xffffffffu;
        #pragma unroll
        for (int b = 0; b < 6; ++b) {   // e in [0,64): 6-bit match-any
          const unsigned vote = (unsigned)__ballot((e >> b) & 1);
          mask &= ((e >> b) & 1) ? vote : ~vote;
        }
        const int leader = __ffs(mask) - 1;
        const int cntg   = __popc(mask);
        const int rank   = __popc(mask & lt);     // stable: rank by lane id
        int bse = 0;
        if (lane == leader) bse = atomicAdd(&offs[wave][e], cntg);
        bse = __shfl(bse, leader, 32);
        sorted[bse + rank] = (int)(base + t * TILE + j * 32 + lane);
      }
    }
  } else {
    // tail chunk: guarded path with 7-bit match-any (sentinel E+64 = 128 has only bit 7 set)
    const int iters = chunk / 32;
    for (int it = 0; it < iters; ++it) {
      const long long idx = base + (long long)it * 32 + lane;
      __builtin_prefetch(&flat[idx + 1024]);
      const int e = (idx < (long long)numel) ? flat[idx] : (E + 64);
      unsigned mask = 0xffffffffu;
      #pragma unroll
      for (int b = 0; b < 7; ++b) {
        const unsigned vote = (unsigned)__ballot((e >> b) & 1);
        mask &= ((e >> b) & 1) ? vote : ~vote;
      }
      const int leader = __ffs(mask) - 1;
      const int cntg   = __popc(mask);
      const int rank   = __popc(mask & lt);
      int bse = 0;
      if (e < E && lane == leader) bse = atomicAdd(&offs[wave][e], cntg);
      bse = __shfl(bse, leader, 32);
      if (e < E) sorted[bse + rank] = (int)idx;
    }
  }
}

// ---------------- Host launcher ----------------
extern "C" void kernel_launch(void* const* d_in, const int* in_sizes, int n_in,
                              void* d_out, int out_size, void* d_ws, size_t ws_size,
                              hipStream_t stream) {
  const int* flat = (const int*)d_in[0];
  const int numel = in_sizes[0];                    // 2097152 * 8

  const int n1 = numel + (E + 1) * (MOE_B - 1);     // max_num_tokens_padded
  const int n2 = n1 / MOE_B;                        // max_num_m_blocks

  int* sorted     = (int*)d_out;
  int* expert_ids = sorted + n1;
  int* total_out  = sorted + n1 + n2;

  // workspace: [E][num_chunks] chunk counts/offsets, then cum[E+1], counts[E]
  const size_t ws_ints = ws_size / sizeof(int);
  long long maxc = (ws_ints > 256) ? (long long)((ws_ints - 160) / E) : 1;
  int num_chunks = (maxc < 4096) ? (int)maxc : 4096;
  if (num_chunks < 1) num_chunks = 1;
  long long chunk_ll = ((long long)numel + num_chunks - 1) / num_chunks;
  chunk_ll = (chunk_ll + 1023) & ~1023LL;           // multiple of WAVES*TILE
  const int chunk = (int)chunk_ll;
  num_chunks = (int)(((long long)numel + chunk - 1) / chunk);

  int* ws         = (int*)d_ws;
  int* chunk_cnts = ws;                              // transposed [E][num_chunks]
  int* cum        = ws + (size_t)E * num_chunks;
  int* counts     = cum + (E + 8);

  k_hist<<<num_chunks, THREADS, 0, stream>>>(flat, numel, chunk, num_chunks, chunk_cnts);
  k_scan_expert<<<E, THREADS, 0, stream>>>(chunk_cnts, num_chunks, counts);
  k_cum<<<1, 128, 0, stream>>>(counts, cum, total_out);
  k_fill<<<E + 1, 256, 0, stream>>>(sorted, cum, counts, numel, n1);
  k_expert_ids<<<(n2 + 255) / 256, 256, 0, stream>>>(expert_ids, cum, n2);
  k_scatter<<<(num_chunks + WAVES - 1) / WAVES, THREADS, 0, stream>>>(
      flat, numel, chunk, num_chunks, chunk_cnts, cum, sorted);
}